// ReplaceableGCNConv_1382979469688
// MI455X (gfx1250) — compile-verified
//
#include <hip/hip_runtime.h>

typedef __attribute__((ext_vector_type(2))) float v2f;
typedef __attribute__((ext_vector_type(4))) float v4f;
typedef __attribute__((ext_vector_type(8))) float v8f;

#define N_NODES 100000
#define N_EDGES 3200000
#define IN_DIM  256
#define OUT_DIM 256
#define EDGES_PER_WAVE 256

// ---------------------------------------------------------------------------
// Kernel 0: pack W (K-major [256][256]) into K-pair-interleaved float2 so a
// B fragment (W[k][n], W[k+1][n]) is one contiguous 8-byte load.
// pw[p*256 + n] = { W[2p][n], W[2p+1][n] },  p = 0..127
// ---------------------------------------------------------------------------
__global__ __launch_bounds__(256)
void pack_w_kernel(const float* __restrict__ w, v2f* __restrict__ pw)
{
    const int idx = blockIdx.x * 256 + threadIdx.x;   // 32768 total
    const int p = idx >> 8;
    const int n = idx & 255;
    v2f t;
    t.x = w[(size_t)(2 * p + 0) * OUT_DIM + n];
    t.y = w[(size_t)(2 * p + 1) * OUT_DIM + n];
    pw[idx] = t;
}

// ---------------------------------------------------------------------------
// Kernel 1: h = x @ W via V_WMMA_F32_16X16X4_F32 (full f32 precision).
// Block = 128 threads = 4 waves; wave w owns a 16(M) x 64(N) strip = 4 tiles.
// x loads are non-temporal: read-once stream, keep L2 free for h (SpMM needs
// h L2-resident: 102.4 MB < 192 MB).
// ---------------------------------------------------------------------------
__global__ __launch_bounds__(128)
void gcn_gemm_wmma(const float* __restrict__ x,
                   const v2f* __restrict__ pw,
                   float* __restrict__ h)
{
    const int lane = threadIdx.x & 31;
    const int wave = threadIdx.x >> 5;
    const int half = lane >> 4;      // 0: lanes 0-15, 1: lanes 16-31
    const int l15  = lane & 15;
    const int m0   = blockIdx.x * 16;
    const int nb   = wave * 64;

    v8f acc0 = {}, acc1 = {}, acc2 = {}, acc3 = {};

    const float* __restrict__ xrow = x + (size_t)(m0 + l15) * IN_DIM;

    for (int k0 = 0; k0 < IN_DIM; k0 += 4) {
        // A fragment: K = {k0+2h, k0+2h+1} for this lane's row (8B NT load)
        const v2f a = __builtin_nontemporal_load(
            (const v2f*)(xrow + k0 + 2 * half));

        // B fragments: packed pair index = k0/2 + half, one b64 per subtile
        const v2f* __restrict__ bp = pw + ((k0 >> 1) + half) * OUT_DIM + nb + l15;
        const v2f b0 = bp[0];
        const v2f b1 = bp[16];
        const v2f b2 = bp[32];
        const v2f b3 = bp[48];

        acc0 = __builtin_amdgcn_wmma_f32_16x16x4_f32(false, a, false, b0, (short)0, acc0, false, false);
        acc1 = __builtin_amdgcn_wmma_f32_16x16x4_f32(false, a, false, b1, (short)0, acc1, false, false);
        acc2 = __builtin_amdgcn_wmma_f32_16x16x4_f32(false, a, false, b2, (short)0, acc2, false, false);
        acc3 = __builtin_amdgcn_wmma_f32_16x16x4_f32(false, a, false, b3, (short)0, acc3, false, false);
    }

    // Store: accumulator VGPR j -> row m0 + j + 8*half (regular stores: h
    // should stay in L2 for the SpMM gather phase).
    float* __restrict__ hb = h + (size_t)(m0 + 8 * half) * OUT_DIM + nb + l15;
    #pragma unroll
    for (int j = 0; j < 8; ++j) {
        hb[(size_t)j * OUT_DIM +  0] = acc0[j];
        hb[(size_t)j * OUT_DIM + 16] = acc1[j];
        hb[(size_t)j * OUT_DIM + 32] = acc2[j];
        hb[(size_t)j * OUT_DIM + 48] = acc3[j];
    }
}

// ---------------------------------------------------------------------------
// Kernel 2: SpMM over sorted COO. One wave per 256-edge chunk.
// Lane l owns columns [4l, 4l+3] and [128+4l, 128+4l+3] -> two b128 gathers
// per edge. Edge metadata loaded coalesced (32 edges, one per lane), then
// broadcast with v_readlane (uniform index) so row/col/value are SGPRs:
// scalar row-boundary compare, scalar gather base address (saddr loads).
// Register accumulation over sorted-row runs; global f32 atomics only at
// row boundaries.
// ---------------------------------------------------------------------------
__device__ __forceinline__ int bcast_i(int v, int j)
{
    return __builtin_amdgcn_readlane(v, j);       // v_readlane_b32 -> SGPR
}
__device__ __forceinline__ float bcast_f(float v, int j)
{
    return __int_as_float(__builtin_amdgcn_readlane(__float_as_int(v), j));
}

__device__ __forceinline__
void flush_acc(float* __restrict__ out, int r, int lane, v4f& a0, v4f& a1)
{
    float* __restrict__ op = out + (size_t)r * OUT_DIM + lane * 4;
    unsafeAtomicAdd(op + 0,       a0.x);
    unsafeAtomicAdd(op + 1,       a0.y);
    unsafeAtomicAdd(op + 2,       a0.z);
    unsafeAtomicAdd(op + 3,       a0.w);
    unsafeAtomicAdd(op + 128 + 0, a1.x);
    unsafeAtomicAdd(op + 128 + 1, a1.y);
    unsafeAtomicAdd(op + 128 + 2, a1.z);
    unsafeAtomicAdd(op + 128 + 3, a1.w);
    a0 = (v4f){0.f, 0.f, 0.f, 0.f};
    a1 = (v4f){0.f, 0.f, 0.f, 0.f};
}

__global__ __launch_bounds__(128)
void gcn_spmm(const float* __restrict__ h,
              const float* __restrict__ values,
              const int* __restrict__ row,
              const int* __restrict__ col,
              float* __restrict__ out)
{
    const int lane  = threadIdx.x & 31;
    const int wave  = threadIdx.x >> 5;
    const int gwave = blockIdx.x * 4 + wave;

    const int e0 = gwave * EDGES_PER_WAVE;
    if (e0 >= N_EDGES) return;                     // wave-uniform exit
    int e1 = e0 + EDGES_PER_WAVE;
    if (e1 > N_EDGES) e1 = N_EDGES;

    v4f acc0 = {0.f, 0.f, 0.f, 0.f};
    v4f acc1 = {0.f, 0.f, 0.f, 0.f};
    int cur = row[e0];

    for (int base = e0; base < e1; base += 32) {
        const int cnt = min(32, e1 - base);
        const int ee  = min(base + lane, N_EDGES - 1);
        // coalesced metadata for 32 edges (one edge per lane)
        const int   r_l = row[ee];
        const int   c_l = col[ee];
        const float v_l = values[ee];

        for (int j = 0; j < cnt; ++j) {
            const int   r = bcast_i(r_l, j);       // SGPR
            const int   c = bcast_i(c_l, j);       // SGPR
            const float v = bcast_f(v_l, j);       // SGPR

            if (r != cur) {                        // scalar compare/branch
                flush_acc(out, cur, lane, acc0, acc1);
                cur = r;
            }
            // scalar row base + per-lane offset -> saddr-form b128 gathers
            const v4f* __restrict__ hp = (const v4f*)(h + (size_t)c * OUT_DIM);
            const v4f ha = hp[lane];               // global_load_b128
            const v4f hb = hp[lane + 32];          // global_load_b128
            acc0 += v * ha;
            acc1 += v * hb;
        }
    }
    flush_acc(out, cur, lane, acc0, acc1);
}

// ---------------------------------------------------------------------------
extern "C" void kernel_launch(void* const* d_in, const int* in_sizes, int n_in,
                              void* d_out, int out_size, void* d_ws, size_t ws_size,
                              hipStream_t stream)
{
    const float* x      = (const float*)d_in[0];
    const float* weight = (const float*)d_in[1];
    const float* values = (const float*)d_in[2];
    const int*   row    = (const int*)d_in[3];
    const int*   col    = (const int*)d_in[4];
    float*       out    = (float*)d_out;

    // Workspace layout: h (102.4 MB) then packed W (256 KB)
    float* h  = (float*)d_ws;
    v2f*   pw = (v2f*)(h + (size_t)N_NODES * OUT_DIM);

    // Harness poisons d_out; atomic accumulation needs zeros.
    hipMemsetAsync(out, 0, (size_t)N_NODES * OUT_DIM * sizeof(float), stream);

    // Pack W: 128*256 threads cover 32768 float2 elements.
    pack_w_kernel<<<dim3(128), 256, 0, stream>>>(weight, pw);

    // GEMM: 100000/16 = 6250 row-tiles, 4 waves (full N=256) per block.
    gcn_gemm_wmma<<<dim3(N_NODES / 16), 128, 0, stream>>>(x, pw, h);

    // SpMM: 3.2M edges / 256 per wave = 12500 waves, 4 waves per block.
    const int nwaves  = (N_EDGES + EDGES_PER_WAVE - 1) / EDGES_PER_WAVE;
    const int nblocks = (nwaves + 3) / 4;
    gcn_spmm<<<nblocks, 128, 0, stream>>>(h, values, row, col, out);
}